// Elastic_block_50276887166995
// MI455X (gfx1250) — compile-verified
//
#include <hip/hip_runtime.h>
#include <math.h>

#define DEV __device__ __forceinline__

// ---------------------------------------------------------------------------
// Complex helpers (double) for the one-time constant build
// ---------------------------------------------------------------------------
struct cplx { double re, im; };
DEV cplx cmul(cplx a, cplx b){ return cplx{a.re*b.re - a.im*b.im, a.re*b.im + a.im*b.re}; }
DEV cplx cadd(cplx a, cplx b){ return cplx{a.re + b.re, a.im + b.im}; }
DEV cplx cscale(cplx a, double s){ return cplx{a.re*s, a.im*s}; }

DEV double factd(int n){ double r = 1.0; for (int i = 2; i <= n; ++i) r *= (double)i; return r; }

DEV double cgc(int j1,int m1,int j2,int m2,int j3,int m3){
  if (m3 != m1 + m2) return 0.0;
  double pref = (double)(2*j3+1)
      * factd(j3+j1-j2) * factd(j3-j1+j2) * factd(j1+j2-j3) / factd(j1+j2+j3+1);
  pref *= factd(j3+m3)*factd(j3-m3)*factd(j1-m1)*factd(j1+m1)*factd(j2-m2)*factd(j2+m2);
  int vmin = 0;
  if (j2-j3-m1 > vmin) vmin = j2-j3-m1;
  if (j1-j3+m2 > vmin) vmin = j1-j3+m2;
  int vmax = j1+j2-j3;
  if (j1-m1 < vmax) vmax = j1-m1;
  if (j2+m2 < vmax) vmax = j2+m2;
  double s = 0.0;
  for (int v = vmin; v <= vmax; ++v){
    double d = factd(v)*factd(j1+j2-j3-v)*factd(j1-m1-v)*factd(j2+m2-v)
             * factd(j3-j2+m1+v)*factd(j3-j1-m2+v);
    s += ((v & 1) ? -1.0 : 1.0) / d;
  }
  return sqrt(pref) * s;
}

DEV void qmat(int l, cplx q[7][7]){
  const int d = 2*l + 1;
  for (int i = 0; i < d; ++i)
    for (int j = 0; j < d; ++j) q[i][j] = cplx{0.0, 0.0};
  const double r2 = 0.70710678118654752440;
  for (int m = -l; m < 0; ++m){
    q[l+m][l-m] = cplx{ r2, 0.0};   // q[l+m, l+|m|]
    q[l+m][l+m] = cplx{0.0, -r2};   // q[l+m, l-|m|]
  }
  q[l][l] = cplx{1.0, 0.0};
  for (int m = 1; m <= l; ++m){
    double s = (m & 1) ? -1.0 : 1.0;
    q[l+m][l+m] = cplx{s*r2, 0.0};
    q[l+m][l-m] = cplx{0.0, s*r2};
  }
  cplx ph;                                  // (-i)^l
  switch (l & 3){
    case 0:  ph = cplx{ 1.0,  0.0}; break;
    case 1:  ph = cplx{ 0.0, -1.0}; break;
    case 2:  ph = cplx{-1.0,  0.0}; break;
    default: ph = cplx{ 0.0,  1.0}; break;
  }
  for (int i = 0; i < d; ++i)
    for (int j = 0; j < d; ++j) q[i][j] = cmul(ph, q[i][j]);
}

// Wigner 3j tensor, staged contraction.  Max dims: d1<=7, d2<=5, d3<=5.
DEV void w3j(int l1, int l2, int l3, double out[7][5][5]){
  const int d1 = 2*l1+1, d2 = 2*l2+1, d3 = 2*l3+1;
  double C[7][5][5];
  for (int a = 0; a < d1; ++a)
    for (int b = 0; b < d2; ++b)
      for (int c = 0; c < d3; ++c) C[a][b][c] = 0.0;
  for (int m1 = -l1; m1 <= l1; ++m1)
    for (int m2 = -l2; m2 <= l2; ++m2){
      int m3 = m1 + m2;
      if (m3 >= -l3 && m3 <= l3)
        C[l1+m1][l2+m2][l3+m3] = cgc(l1, m1, l2, m2, l3, m3);
    }
  cplx q1[7][7], q2[7][7], q3[7][7];
  qmat(l1, q1); qmat(l2, q2); qmat(l3, q3);

  cplx t1[7][5][5];
  for (int a = 0; a < d1; ++a)
    for (int m = 0; m < d2; ++m)
      for (int n = 0; n < d3; ++n){
        cplx s = cplx{0.0, 0.0};
        for (int l = 0; l < d1; ++l) s = cadd(s, cscale(q1[a][l], C[l][m][n]));
        t1[a][m][n] = s;
      }
  cplx t2[7][5][5];
  for (int a = 0; a < d1; ++a)
    for (int b = 0; b < d2; ++b)
      for (int n = 0; n < d3; ++n){
        cplx s = cplx{0.0, 0.0};
        for (int m = 0; m < d2; ++m) s = cadd(s, cmul(q2[b][m], t1[a][m][n]));
        t2[a][b][n] = s;
      }
  double wr[7][5][5], wi[7][5][5];
  double nr = 0.0, ni = 0.0;
  for (int a = 0; a < d1; ++a)
    for (int b = 0; b < d2; ++b)
      for (int c = 0; c < d3; ++c){
        cplx s = cplx{0.0, 0.0};
        for (int n = 0; n < d3; ++n){
          cplx q3c = cplx{q3[c][n].re, -q3[c][n].im};
          s = cadd(s, cmul(q3c, t2[a][b][n]));
        }
        wr[a][b][c] = s.re; wi[a][b][c] = s.im;
        nr += s.re*s.re;    ni += s.im*s.im;
      }
  const bool useR = (nr >= ni);
  const double inv = 1.0 / sqrt(useR ? nr : ni);
  for (int a = 0; a < d1; ++a)
    for (int b = 0; b < d2; ++b)
      for (int c = 0; c < d3; ++c)
        out[a][b][c] = (useR ? wr[a][b][c] : wi[a][b][c]) * inv;
}

__device__ __constant__ int PL[18][3] = {
  {0,0,0},{0,1,1},{0,2,2},{1,0,1},{1,1,0},{1,1,1},{1,1,2},{1,2,1},{1,2,2},
  {2,0,2},{2,1,1},{2,1,2},{2,2,0},{2,2,1},{2,2,2},{3,1,2},{3,2,1},{3,2,2}};
__device__ __constant__ int L1OFF[4] = {0,2,8,18};
__device__ __constant__ int IROFF[3] = {0,1,4};
__device__ __constant__ int NOUT[3]  = {3,7,8};
__device__ __constant__ int IDX6[6]  = {0,4,8,1,5,6};

// ---------------------------------------------------------------------------
// Prep kernel: build AT[48][32] = (padded, transposed) fused linear map so that
//   out[b, j] = sum_i X[b,i] * AT[j][i]          (j < 36; cols 36..47 = 0)
// One wave: lanes 0..2 build Q_CART rows in LDS, lanes 0..17 handle one path.
// ---------------------------------------------------------------------------
__global__ void prep_kernel(const float* __restrict__ wt, float* __restrict__ AT){
  const int t = (int)threadIdx.x;
  __shared__ double QC[9][9];
  for (int i = t; i < 48*32; i += (int)blockDim.x) AT[i] = 0.0f;
  if (t < 3){
    double w[7][5][5];
    w3j(1, 1, t, w);
    const int dd = 2*t + 1;
    const double sc = sqrt((double)(2*t + 1));
    for (int n = 0; n < dd; ++n)
      for (int a = 0; a < 3; ++a)
        for (int b = 0; b < 3; ++b)
          QC[IROFF[t] + n][a*3 + b] = sc * w[a][b][n];
  }
  __syncthreads();
  if (t < 18){
    const int l1 = PL[t][0], l2 = PL[t][1], lo = PL[t][2];
    const int d1 = 2*l1+1, d2 = 2*l2+1, dd = 2*lo+1;
    double w[7][5][5];
    w3j(l1, l2, lo, w);
    const double wp = (double)wt[t] * sqrt((double)dd / (double)NOUT[lo]);
    for (int i = 0; i < d1; ++i)
      for (int a = 0; a < 6; ++a)
        for (int c = 0; c < 6; ++c){
          double s = 0.0;
          for (int k = 0; k < dd; ++k)
            for (int j = 0; j < d2; ++j)
              s += QC[IROFF[lo] + k][IDX6[a]] * w[i][j][k] * QC[IROFF[l2] + j][IDX6[c]];
          const float v = (float)(wp * s);
          if (v != 0.0f)
            atomicAdd(&AT[(a*6 + c)*32 + (L1OFF[l1] + i)], v);
        }
  }
}

// ---------------------------------------------------------------------------
// GEMM: (rows x 32) @ (32 x 48pad) via V_WMMA_F32_16X16X4_F32, one 16-row tile
// per wave iteration, 8 K-steps x 3 N-tiles = 24 WMMAs per tile.
// ---------------------------------------------------------------------------
typedef __attribute__((ext_vector_type(2))) float v2f;
typedef __attribute__((ext_vector_type(8))) float v8f;

__global__ void __launch_bounds__(256) gemm_kernel(
    const float* __restrict__ X, const float* __restrict__ AT,
    float* __restrict__ Y, int ntiles){
  const int lane = (int)(threadIdx.x & 31u);
  const int m = lane & 15;        // column-in-tile / row-in-tile lane id
  const int h = lane >> 4;        // lane half
  const int wid = (int)((blockIdx.x * blockDim.x + threadIdx.x) >> 5);
  const int nw  = (int)((gridDim.x * blockDim.x) >> 5);

  // B fragments: AT is [col][k] so lane(m,h) K-pair (4t+2h, 4t+2h+1) is one b64.
  v2f b[8][3];
#pragma unroll
  for (int t = 0; t < 8; ++t)
#pragma unroll
    for (int n = 0; n < 3; ++n)
      b[t][n] = *(const v2f*)(AT + (16*n + m)*32 + 4*t + 2*h);

  for (int tile = wid; tile < ntiles; tile += nw){
    const float* rp = X + ((size_t)tile*16 + (size_t)m)*32 + 2*h;
    if (tile + nw < ntiles)
      __builtin_prefetch(X + ((size_t)(tile + nw)*16 + (size_t)m)*32 + 2*h, 0, 1);
    v8f c0 = {}, c1 = {}, c2 = {};
#pragma unroll
    for (int t = 0; t < 8; ++t){
      v2f a = *(const v2f*)(rp + 4*t);   // A fragment: K = 4t + 2h + {0,1}
      c0 = __builtin_amdgcn_wmma_f32_16x16x4_f32(false, a, false, b[t][0], (short)0, c0, false, false);
      c1 = __builtin_amdgcn_wmma_f32_16x16x4_f32(false, a, false, b[t][1], (short)0, c1, false, false);
      c2 = __builtin_amdgcn_wmma_f32_16x16x4_f32(false, a, false, b[t][2], (short)0, c2, false, false);
    }
    // D layout: VGPR r, lane -> row = r + 8h, col = m (per N-tile).
    float* yp = Y + ((size_t)tile*16 + (size_t)(8*h))*36 + m;
#pragma unroll
    for (int r = 0; r < 8; ++r){         // unconditional stores: cols m, 16+m
      yp[(size_t)r*36]      = c0[r];
      yp[(size_t)r*36 + 16] = c1[r];
    }
    if (m < 4){                          // single exec-mask toggle per tile
#pragma unroll
      for (int r = 0; r < 8; ++r)
        yp[(size_t)r*36 + 32] = c2[r];   // cols 32..35 only (36-wide output)
    }
  }
}

// Scalar tail for rows not covered by 16-row tiles (none for bs = 1e6).
__global__ void tail_kernel(const float* __restrict__ X, const float* __restrict__ AT,
                            float* __restrict__ Y, int start, int nrows){
  int idx = (int)(blockIdx.x * blockDim.x + threadIdx.x);
  int total = (nrows - start) * 36;
  if (idx >= total) return;
  int row = start + idx / 36, col = idx % 36;
  float s = 0.0f;
  for (int i = 0; i < 32; ++i) s += X[(size_t)row*32 + i] * AT[col*32 + i];
  Y[(size_t)row*36 + col] = s;
}

extern "C" void kernel_launch(void* const* d_in, const int* in_sizes, int n_in,
                              void* d_out, int out_size, void* d_ws, size_t ws_size,
                              hipStream_t stream){
  (void)n_in; (void)out_size; (void)ws_size;
  const float* X  = (const float*)d_in[0];
  const float* wt = (const float*)d_in[1];
  float* Y  = (float*)d_out;
  float* AT = (float*)d_ws;                 // 48*32 floats = 6 KB scratch

  const int nrows  = in_sizes[0] / 32;
  const int ntiles = nrows / 16;

  prep_kernel<<<1, 32, 0, stream>>>(wt, AT);

  if (ntiles > 0){
    int blocks = (ntiles + 7) / 8;          // 8 waves per block
    if (blocks > 2048) blocks = 2048;
    gemm_kernel<<<blocks, 256, 0, stream>>>(X, AT, Y, ntiles);
  }
  const int tail = nrows - ntiles * 16;
  if (tail > 0){
    int tot = tail * 36;
    tail_kernel<<<(tot + 255) / 256, 256, 0, stream>>>(X, AT, Y, ntiles * 16, nrows);
  }
}